// GCNII_model_62431644615048
// MI455X (gfx1250) — compile-verified
//
#include <hip/hip_runtime.h>
#include <cmath>

#define NN 100000
#define EE 1600000
#define FF 512
#define HH 64
#define CC 40
#define LL 16

typedef __attribute__((ext_vector_type(2))) float v2f;
typedef __attribute__((ext_vector_type(8))) float v8f;

// ---------------- degree / norm ----------------
__global__ void deg_init_kernel(float* __restrict__ deg) {
    int i = blockIdx.x * 256 + threadIdx.x;
    if (i < NN) deg[i] = 1.0f;   // self-loop contributes 1 to deg[src]
}

__global__ void deg_count_kernel(const long long* __restrict__ ei, float* __restrict__ deg) {
    int e = blockIdx.x * 256 + threadIdx.x;
    if (e < EE) atomicAdd(&deg[(int)ei[e]], 1.0f);   // ei[0:E] = src
}

__global__ void dinv_kernel(float* __restrict__ deg) {
    int i = blockIdx.x * 256 + threadIdx.x;
    if (i < NN) deg[i] = rsqrtf(deg[i]);   // deg >= 1 always (self-loops)
}

__global__ void zero_kernel(float* __restrict__ p, int n) {
    int i = blockIdx.x * 256 + threadIdx.x;
    if (i < n) p[i] = 0.0f;
}

// ---------------- input projection: h = relu(x @ w0 + b0) ----------------
// block = 128 threads (4 waves), computes 16 rows x 64 cols; wave w owns col tile w.
__global__ __launch_bounds__(128) void proj_kernel(const float* __restrict__ x,
                                                   const float* __restrict__ w0,
                                                   const float* __restrict__ b0,
                                                   float* __restrict__ h,
                                                   float* __restrict__ h0) {
    __shared__ float As[16][65];
    __shared__ float Bs[64][65];
    const int tid  = threadIdx.x;
    const int row0 = blockIdx.x * 16;
    const int wave = tid >> 5;
    const int lane = tid & 31;
    const int l15  = lane & 15;
    const int koff = (lane >> 4) * 2;   // lanes 0-15: K+0/K+1, lanes 16-31: K+2/K+3
    const int col0 = wave * 16;

    v8f acc = {};
    for (int k0 = 0; k0 < FF; k0 += 64) {
        for (int i = tid; i < 16 * 64; i += 128) {
            int r = i >> 6, c = i & 63;
            As[r][c] = x[(size_t)(row0 + r) * FF + (k0 + c)];
        }
        for (int i = tid; i < 64 * 64; i += 128) {
            int r = i >> 6, c = i & 63;
            Bs[r][c] = w0[(size_t)(k0 + r) * HH + c];
        }
        __syncthreads();
        #pragma unroll
        for (int kk = 0; kk < 64; kk += 4) {
            v2f a, b;
            a.x = As[l15][kk + koff];
            a.y = As[l15][kk + koff + 1];
            b.x = Bs[kk + koff][col0 + l15];
            b.y = Bs[kk + koff + 1][col0 + l15];
            acc = __builtin_amdgcn_wmma_f32_16x16x4_f32(false, a, false, b,
                                                        (short)0, acc, false, false);
        }
        __syncthreads();
    }
    #pragma unroll
    for (int r = 0; r < 8; r++) {
        int mrow = (lane < 16) ? r : (r + 8);
        int col  = col0 + l15;
        float v = acc[r] + b0[col];
        v = v > 0.0f ? v : 0.0f;
        size_t idx = (size_t)(row0 + mrow) * HH + col;
        h[idx]  = v;
        h0[idx] = v;
    }
}

// ---------------- sparse aggregation: agg[dst] += dinv[s]*dinv[d] * h[src] ----------------
// 64 consecutive lanes = 64 features of one edge -> coalesced gather + coalesced atomics.
__global__ __launch_bounds__(256) void scatter_kernel(const long long* __restrict__ ei,
                                                      const float* __restrict__ dinv,
                                                      const float* __restrict__ h,
                                                      float* __restrict__ agg) {
    long long g = (long long)blockIdx.x * 256 + threadIdx.x;
    int e = (int)(g >> 6);
    int f = (int)(g & 63);
    if (e < EE) {
        int s = (int)ei[e];
        int d = (int)ei[EE + e];
        float w = dinv[s] * dinv[d];
        atomicAdd(&agg[(size_t)d * HH + f], w * h[(size_t)s * HH + f]);
    }
}

// ---------------- GCNII layer: h = relu(beta*(S @ W) + (1-beta)*S) ----------------
// S = 0.9*(agg + dinv^2*h) + 0.1*h0  (self-loop folded in). In-place h update.
__global__ __launch_bounds__(128) void layer_kernel(float* __restrict__ h,
                                                    const float* __restrict__ h0,
                                                    const float* __restrict__ agg,
                                                    const float* __restrict__ dinv,
                                                    const float* __restrict__ W,
                                                    float beta) {
    __shared__ float Ss[16][65];
    __shared__ float Ws[64][65];
    const int tid  = threadIdx.x;
    const int row0 = blockIdx.x * 16;

    for (int i = tid; i < 16 * 64; i += 128) {
        int r = i >> 6, c = i & 63;
        int gr = row0 + r;
        size_t idx = (size_t)gr * HH + c;
        float di = dinv[gr];
        Ss[r][c] = 0.9f * (agg[idx] + di * di * h[idx]) + 0.1f * h0[idx];
    }
    for (int i = tid; i < 64 * 64; i += 128) {
        int r = i >> 6, c = i & 63;
        Ws[r][c] = W[r * HH + c];
    }
    __syncthreads();

    const int wave = tid >> 5;
    const int lane = tid & 31;
    const int l15  = lane & 15;
    const int koff = (lane >> 4) * 2;
    const int col0 = wave * 16;

    v8f acc = {};
    #pragma unroll
    for (int kk = 0; kk < 64; kk += 4) {
        v2f a, b;
        a.x = Ss[l15][kk + koff];
        a.y = Ss[l15][kk + koff + 1];
        b.x = Ws[kk + koff][col0 + l15];
        b.y = Ws[kk + koff + 1][col0 + l15];
        acc = __builtin_amdgcn_wmma_f32_16x16x4_f32(false, a, false, b,
                                                    (short)0, acc, false, false);
    }
    const float ib = 1.0f - beta;
    #pragma unroll
    for (int r = 0; r < 8; r++) {
        int mrow = (lane < 16) ? r : (r + 8);
        int col  = col0 + l15;
        float v = beta * acc[r] + ib * Ss[mrow][col];
        v = v > 0.0f ? v : 0.0f;
        h[(size_t)(row0 + mrow) * HH + col] = v;
    }
}

// ---------------- classifier + log_softmax ----------------
__global__ __launch_bounds__(64) void cls_kernel(const float* __restrict__ h,
                                                 const float* __restrict__ wf,
                                                 const float* __restrict__ bf,
                                                 float* __restrict__ out) {
    __shared__ float hrow[64];
    __shared__ float logits[CC];
    __shared__ float lse;
    int row = blockIdx.x;
    int t   = threadIdx.x;
    hrow[t] = h[(size_t)row * HH + t];
    __syncthreads();
    if (t < CC) {
        float s = bf[t];
        #pragma unroll
        for (int k = 0; k < HH; k++) s += hrow[k] * wf[k * CC + t];
        logits[t] = s;
    }
    __syncthreads();
    if (t == 0) {
        float m = logits[0];
        for (int c = 1; c < CC; c++) m = fmaxf(m, logits[c]);
        float s = 0.0f;
        for (int c = 0; c < CC; c++) s += expf(logits[c] - m);
        lse = m + logf(s);
    }
    __syncthreads();
    if (t < CC) out[(size_t)row * CC + t] = logits[t] - lse;
}

// ---------------- launch ----------------
extern "C" void kernel_launch(void* const* d_in, const int* in_sizes, int n_in,
                              void* d_out, int out_size, void* d_ws, size_t ws_size,
                              hipStream_t stream) {
    const float*     x  = (const float*)d_in[0];
    const long long* ei = (const long long*)d_in[1];   // int64 [2, E]
    const float*     w0 = (const float*)d_in[2];
    const float*     b0 = (const float*)d_in[3];
    const float*     wl = (const float*)d_in[4];       // [L, H, H]
    const float*     wf = (const float*)d_in[5];
    const float*     bf = (const float*)d_in[6];
    float*           out = (float*)d_out;

    char* ws = (char*)d_ws;
    size_t off = 0;
    auto take = [&](size_t bytes) -> void* {
        void* p = ws + off;
        off = (off + bytes + 255) & ~(size_t)255;
        return p;
    };
    float* dinv = (float*)take((size_t)NN * sizeof(float));        // deg then dinv (in place)
    float* h    = (float*)take((size_t)NN * HH * sizeof(float));
    float* h0   = (float*)take((size_t)NN * HH * sizeof(float));
    float* agg  = (float*)take((size_t)NN * HH * sizeof(float));
    (void)ws_size; (void)in_sizes; (void)n_in; (void)out_size;

    deg_init_kernel<<<(NN + 255) / 256, 256, 0, stream>>>(dinv);
    deg_count_kernel<<<(EE + 255) / 256, 256, 0, stream>>>(ei, dinv);
    dinv_kernel<<<(NN + 255) / 256, 256, 0, stream>>>(dinv);

    proj_kernel<<<NN / 16, 128, 0, stream>>>(x, w0, b0, h, h0);

    for (int i = 0; i < LL; i++) {
        float beta = logf(0.5f / (float)(i + 1) + 1.0f);
        zero_kernel<<<(NN * HH + 255) / 256, 256, 0, stream>>>(agg, NN * HH);
        scatter_kernel<<<(EE * 64) / 256, 256, 0, stream>>>(ei, dinv, h, agg);
        layer_kernel<<<NN / 16, 128, 0, stream>>>(h, h0, agg, dinv,
                                                  wl + (size_t)i * HH * HH, beta);
    }

    cls_kernel<<<NN, 64, 0, stream>>>(h, wf, bf, out);
}